// InputEmbedding_8890582303480
// MI455X (gfx1250) — compile-verified
//
#include <hip/hip_runtime.h>
#include <stdint.h>

// Shapes (compile-time constants from the reference)
#define Bsz 64
#define Tlen 288
#define Vn 24
#define Dd 256
#define TILE_T 8
#define NTHREADS 256
#define VD4 ((Vn * Dd) / 4)   // 1536 float4 per timestep

typedef __attribute__((ext_vector_type(4))) float v4f;

// ---- CDNA5 async global->LDS copies (ASYNCcnt-tracked, per-lane) ----------
// the LDS address is taken from the VDST VGPR; generic pointers to __shared__ carry the
// LDS byte offset in their low 32 bits (aperture form), so truncation gives
// the required VGPR value.
__device__ __forceinline__ void async_ld_b32(void* lds_dst, const void* gsrc) {
    unsigned off = (unsigned)(uintptr_t)lds_dst;
    asm volatile("global_load_async_to_lds_b32 %0, %1, off"
                 :: "v"(off), "v"(gsrc) : "memory");
}
__device__ __forceinline__ void async_ld_b128(void* lds_dst, const void* gsrc) {
    unsigned off = (unsigned)(uintptr_t)lds_dst;
    asm volatile("global_load_async_to_lds_b128 %0, %1, off"
                 :: "v"(off), "v"(gsrc) : "memory");
}
__device__ __forceinline__ void wait_async_zero() {
    asm volatile("s_wait_asynccnt 0x0" ::: "memory");
}

__global__ __launch_bounds__(NTHREADS)
void input_embedding_kernel(const float* __restrict__ features,
                            const int*   __restrict__ bar_in_day,
                            const int*   __restrict__ day_of_week,
                            const int*   __restrict__ day_of_month,
                            const int*   __restrict__ day_of_year,
                            const float* __restrict__ W,
                            const float* __restrict__ bias,
                            const float* __restrict__ tod_table,
                            const float* __restrict__ dow_table,
                            const float* __restrict__ dom_table,
                            const float* __restrict__ doy_table,
                            float*       __restrict__ out)
{
    __shared__ __align__(16) float sW[Vn * Dd];        // 24 KB
    __shared__ __align__(16) float sB[Vn * Dd];        // 24 KB
    __shared__ __align__(16) float sPos[TILE_T * Dd];  // 8 KB
    __shared__ __align__(16) float sX[TILE_T * Vn];    // 768 B

    const int tid = threadIdx.x;
    const int b   = blockIdx.y;
    const int t0  = blockIdx.x * TILE_T;

    // ---- stage W and bias (float4 granularity) via async b128 ----
    for (int j = tid; j < VD4; j += NTHREADS) {
        async_ld_b128(&sW[j * 4], W    + (size_t)j * 4);
        async_ld_b128(&sB[j * 4], bias + (size_t)j * 4);
    }

    // ---- stage the 8x24 feature tile ----
    if (tid < TILE_T * Vn) {
        const int tl = tid / Vn;
        const int v  = tid % Vn;
        async_ld_b32(&sX[tid],
                     features + ((size_t)b * Tlen + (t0 + tl)) * Vn + v);
    }

    // ---- positional gather: thread tid handles channel d = tid ----
    const int dw = day_of_week[b];
    const int dm = day_of_month[b];
    const int dy = day_of_year[b];
    const int d  = tid;
    for (int tl = 0; tl < TILE_T; ++tl) {
        const int tod_i = bar_in_day[b * Tlen + t0 + tl];
        const float* src;
        if (d < 128)      src = tod_table + (size_t)tod_i * 128 + d;
        else if (d < 160) src = dow_table + (size_t)dw * 32 + (d - 128);
        else if (d < 192) src = dom_table + (size_t)dm * 32 + (d - 160);
        else              src = doy_table + (size_t)dy * 64 + (d - 192);
        async_ld_b32(&sPos[tl * Dd + d], src);
    }

    wait_async_zero();   // wave's async LDS writes complete
    __syncthreads();     // visible to whole workgroup

    const v4f* sW4 = (const v4f*)sW;
    const v4f* sB4 = (const v4f*)sB;
    const v4f* sP4 = (const v4f*)sPos;
    v4f* out4 = (v4f*)out + ((size_t)b * Tlen + t0) * VD4;

    // ---- streaming main loop: 6 iters x 8 timesteps of B128 NT stores ----
    for (int j = tid; j < VD4; j += NTHREADS) {
        const int v  = j >> 6;   // 64 float4 per variable row
        const int d4 = j & 63;
        const v4f w4 = sW4[j];
        const v4f b4 = sB4[j];
#pragma unroll
        for (int tl = 0; tl < TILE_T; ++tl) {
            const float x = sX[tl * Vn + v];
            const v4f p   = sP4[tl * 64 + d4];
            v4f o;
            o.x = fmaf(x, w4.x, b4.x + p.x);
            o.y = fmaf(x, w4.y, b4.y + p.y);
            o.z = fmaf(x, w4.z, b4.z + p.z);
            o.w = fmaf(x, w4.w, b4.w + p.w);
            __builtin_nontemporal_store(o, &out4[(size_t)tl * VD4 + j]);
        }
    }
}

extern "C" void kernel_launch(void* const* d_in, const int* in_sizes, int n_in,
                              void* d_out, int out_size, void* d_ws, size_t ws_size,
                              hipStream_t stream) {
    (void)in_sizes; (void)n_in; (void)out_size; (void)d_ws; (void)ws_size;
    const float* features    = (const float*)d_in[0];
    const int*   bar_in_day  = (const int*)  d_in[1];
    const int*   day_of_week = (const int*)  d_in[2];
    const int*   day_of_month= (const int*)  d_in[3];
    const int*   day_of_year = (const int*)  d_in[4];
    const float* W           = (const float*)d_in[5];
    const float* bias        = (const float*)d_in[6];
    const float* tod_table   = (const float*)d_in[7];
    const float* dow_table   = (const float*)d_in[8];
    const float* dom_table   = (const float*)d_in[9];
    const float* doy_table   = (const float*)d_in[10];

    dim3 grid(Tlen / TILE_T, Bsz);   // 36 x 64 = 2304 blocks
    dim3 block(NTHREADS);
    input_embedding_kernel<<<grid, block, 0, stream>>>(
        features, bar_in_day, day_of_week, day_of_month, day_of_year,
        W, bias, tod_table, dow_table, dom_table, doy_table,
        (float*)d_out);
}